// RNNSeq2SeqDecoder_64338610094362
// MI455X (gfx1250) — compile-verified
//
#include <hip/hip_runtime.h>
#include <math.h>

// Problem constants (from reference)
#define Bc 32
#define Tc 16
#define Sc 400
#define Hc 300
#define Ec 300
#define Lc 3
#define Vc 50000
#define OOVc 20
#define Rc (Bc * Tc)        // 512 rows (b,t)
#define VPc (Vc + OOVc)     // 50020 padded vocab
#define NEG_INF_F (-1e9f)

#define KCHUNK 300          // K staging chunk (all our K are multiples of 300)
#define KPAD   304          // LDS row stride (pad to scatter banks)

typedef float v2f __attribute__((ext_vector_type(2)));
typedef float v8f __attribute__((ext_vector_type(8)));

// ---------------------------------------------------------------------------
// K1: embedding gather  x[r,e] = embed[tokens[r], e]
// ---------------------------------------------------------------------------
__global__ void k_embed(const int* __restrict__ tokens,
                        const float* __restrict__ embed,
                        float* __restrict__ x) {
  int idx = blockIdx.x * blockDim.x + threadIdx.x;
  if (idx >= Rc * Ec) return;
  int r = idx / Ec, e = idx % Ec;
  x[idx] = embed[(long)tokens[r] * Ec + e];
}

// ---------------------------------------------------------------------------
// K2: one GRU cell (one timestep, one layer). Block per batch element.
// inp row b at inp + b*inp_stride (length 300). h_in/h_out are [B,H].
// ---------------------------------------------------------------------------
__global__ void k_gru_cell(const float* __restrict__ inp, int inp_stride,
                           const float* __restrict__ h_in,
                           float* __restrict__ h_out,
                           const float* __restrict__ Wih,
                           const float* __restrict__ Whh,
                           const float* __restrict__ bih,
                           const float* __restrict__ bhh,
                           float* __restrict__ hiddens_t) {
  __shared__ float s_in[Hc];
  __shared__ float s_h[Hc];
  int b = blockIdx.x;
  for (int k = threadIdx.x; k < Hc; k += blockDim.x) {
    s_in[k] = inp[(long)b * inp_stride + k];
    s_h[k]  = h_in[b * Hc + k];
  }
  __syncthreads();
  int j = threadIdx.x;
  if (j < Hc) {
    const float* wr = Wih + (long)j * Ec;
    const float* wz = Wih + (long)(Hc + j) * Ec;
    const float* wn = Wih + (long)(2 * Hc + j) * Ec;
    const float* ur = Whh + (long)j * Hc;
    const float* uz = Whh + (long)(Hc + j) * Hc;
    const float* un = Whh + (long)(2 * Hc + j) * Hc;
    float gir = bih[j], giz = bih[Hc + j], gin = bih[2 * Hc + j];
    float ghr = bhh[j], ghz = bhh[Hc + j], ghn = bhh[2 * Hc + j];
    for (int k = 0; k < Hc; ++k) {
      float xi = s_in[k], hi = s_h[k];
      gir = fmaf(wr[k], xi, gir);
      giz = fmaf(wz[k], xi, giz);
      gin = fmaf(wn[k], xi, gin);
      ghr = fmaf(ur[k], hi, ghr);
      ghz = fmaf(uz[k], hi, ghz);
      ghn = fmaf(un[k], hi, ghn);
    }
    float rg = 1.f / (1.f + __expf(-(gir + ghr)));
    float zg = 1.f / (1.f + __expf(-(giz + ghz)));
    float ng = tanhf(gin + rg * ghn);
    float hn = (1.f - zg) * ng + zg * s_h[j];
    h_out[b * Hc + j] = hn;
    if (hiddens_t) hiddens_t[(long)b * Tc * Hc + j] = hn;  // hiddens[b,t,:]
  }
}

// ---------------------------------------------------------------------------
// Generic C[M,N] = A[M,K] * W[N,K]^T + bias   via V_WMMA_F32_16X16X4_F32.
//
// Block (256 thr, 8 waves) computes a 32-row x 256-col super-tile:
//   - A panel (32 x KCHUNK) staged cooperatively in LDS (coalesced global
//     reads once per block, ds_load_b64 reuse by all 8 waves)
//   - each wave owns a 32x32 register tile (2x2 WMMA accumulators) so every
//     A and W fetch feeds two WMMAs (halves L2 traffic on both streams)
//
// Requirements: M % 32 == 0, K % KCHUNK == 0 (holds for K=300 and K=600).
// Lane layouts per ISA 7.12.2:
//   A 16x4 f32: lane L -> m=L%16, half=L/16; VGPRs = A[m][k+2h], A[m][k+2h+1]
//   B 4x16 f32: lane L -> n=L%16;          VGPRs = W[n][k+2h], W[n][k+2h+1]
//   C/D:        VGPR v -> M = v + 8*half, N = lane%16
// ---------------------------------------------------------------------------
__global__ void k_gemm_nt_wmma(const float* __restrict__ A, int lda,
                               const float* __restrict__ W, int ldw,
                               const float* __restrict__ bias,
                               float* __restrict__ C, long ldc,
                               int blocks_m, int N, int K) {
  __shared__ float sA[32 * KPAD];
  int bm = blockIdx.x % blocks_m;   // bm-major: concurrent blocks share W panel
  int bn = blockIdx.x / blocks_m;
  int tid  = threadIdx.x;
  int wave = tid >> 5;
  int lane = tid & 31;
  int lm   = lane & 15;
  int half = lane >> 4;
  int m0    = bm * 32;
  int nbase = bn * 256 + wave * 32;
  int n0 = nbase + lm;
  int n1 = nbase + 16 + lm;
  int n0c = (n0 < N) ? n0 : (N - 1);   // clamped: garbage cols never stored
  int n1c = (n1 < N) ? n1 : (N - 1);
  const float* W0 = W + (long)n0c * ldw + 2 * half;
  const float* W1 = W + (long)n1c * ldw + 2 * half;
  int aoff0 = lm * KPAD + 2 * half;          // m-subtile 0 (rows 0..15)
  int aoff1 = (16 + lm) * KPAD + 2 * half;   // m-subtile 1 (rows 16..31)

  v8f acc00 = {}, acc01 = {}, acc10 = {}, acc11 = {};
  for (int kk = 0; kk < K; kk += KCHUNK) {
    // stage A[32][KCHUNK] into LDS (coalesced along k)
    for (int idx = tid; idx < 32 * KCHUNK; idx += 256) {
      int row = idx / KCHUNK;
      int k   = idx % KCHUNK;
      sA[row * KPAD + k] = A[(long)(m0 + row) * lda + kk + k];
    }
    __syncthreads();
    const float* Wk0 = W0 + kk;
    const float* Wk1 = W1 + kk;
    for (int k = 0; k < KCHUNK; k += 4) {
      v2f a0, a1, b0, b1;
      a0.x = sA[aoff0 + k];
      a0.y = sA[aoff0 + k + 1];
      a1.x = sA[aoff1 + k];
      a1.y = sA[aoff1 + k + 1];
      b0.x = Wk0[k];
      b0.y = Wk0[k + 1];
      b1.x = Wk1[k];
      b1.y = Wk1[k + 1];
      acc00 = __builtin_amdgcn_wmma_f32_16x16x4_f32(false, a0, false, b0,
                                                    (short)0, acc00, false, false);
      acc01 = __builtin_amdgcn_wmma_f32_16x16x4_f32(false, a0, false, b1,
                                                    (short)0, acc01, false, false);
      acc10 = __builtin_amdgcn_wmma_f32_16x16x4_f32(false, a1, false, b0,
                                                    (short)0, acc10, false, false);
      acc11 = __builtin_amdgcn_wmma_f32_16x16x4_f32(false, a1, false, b1,
                                                    (short)0, acc11, false, false);
    }
    __syncthreads();
  }

  float bv0 = (n0 < N && bias) ? bias[n0] : 0.f;
  float bv1 = (n1 < N && bias) ? bias[n1] : 0.f;
  for (int v = 0; v < 8; ++v) {
    int mm0 = m0 + v + 8 * half;        // rows of m-subtile 0
    int mm1 = m0 + 16 + v + 8 * half;   // rows of m-subtile 1
    if (n0 < N) {
      C[(long)mm0 * ldc + n0] = acc00[v] + bv0;
      C[(long)mm1 * ldc + n0] = acc10[v] + bv0;
    }
    if (n1 < N) {
      C[(long)mm0 * ldc + n1] = acc01[v] + bv1;
      C[(long)mm1 * ldc + n1] = acc11[v] + bv1;
    }
  }
}

// ---------------------------------------------------------------------------
// K4: attention for one (b,t) row: scores -> masked softmax -> context.
// Writes attn_dist row into d_out tail; writes [context, hidden] into concat.
// ---------------------------------------------------------------------------
__global__ void k_attn(const float* __restrict__ q,
                       const float* __restrict__ enc,
                       const unsigned char* __restrict__ mask,
                       const float* __restrict__ hiddens,
                       float* __restrict__ attn_out,
                       float* __restrict__ concat) {
  __shared__ float s_q[Hc];
  __shared__ float s_sc[Sc];
  __shared__ float red[256];
  int r = blockIdx.x;
  int b = r / Tc;
  int tid = threadIdx.x;
  for (int k = tid; k < Hc; k += 256) s_q[k] = q[(long)r * Hc + k];
  __syncthreads();
  for (int s = tid; s < Sc; s += 256) {
    const float* e = enc + ((long)b * Sc + s) * Hc;
    float d = 0.f;
    for (int k = 0; k < Hc; ++k) d = fmaf(s_q[k], e[k], d);
    s_sc[s] = mask[b * Sc + s] ? d : NEG_INF_F;
  }
  __syncthreads();
  float mx = -INFINITY;
  for (int s = tid; s < Sc; s += 256) mx = fmaxf(mx, s_sc[s]);
  red[tid] = mx;
  __syncthreads();
  for (int o = 128; o > 0; o >>= 1) {
    if (tid < o) red[tid] = fmaxf(red[tid], red[tid + o]);
    __syncthreads();
  }
  mx = red[0];
  __syncthreads();
  float sm = 0.f;
  for (int s = tid; s < Sc; s += 256) {
    float e0 = __expf(s_sc[s] - mx);
    s_sc[s] = e0;
    sm += e0;
  }
  red[tid] = sm;
  __syncthreads();
  for (int o = 128; o > 0; o >>= 1) {
    if (tid < o) red[tid] += red[tid + o];
    __syncthreads();
  }
  float inv = 1.f / red[0];
  __syncthreads();
  for (int s = tid; s < Sc; s += 256) {
    s_sc[s] *= inv;
    attn_out[(long)r * Sc + s] = s_sc[s];
  }
  __syncthreads();
  for (int k = tid; k < Hc; k += 256) {
    float c = 0.f;
    const float* eb = enc + (long)b * Sc * Hc + k;
    for (int s = 0; s < Sc; ++s) c = fmaf(s_sc[s], eb[(long)s * Hc], c);
    concat[(long)r * 2 * Hc + k]      = c;                         // context
    concat[(long)r * 2 * Hc + Hc + k] = hiddens[(long)r * Hc + k]; // hidden
  }
}

// ---------------------------------------------------------------------------
// K6: p_gen = softmax([context, hidden, x] @ pgen_W.T + pgen_b), per row.
// ---------------------------------------------------------------------------
__global__ void k_pgen(const float* __restrict__ concat,
                       const float* __restrict__ x,
                       const float* __restrict__ pw,
                       const float* __restrict__ pb,
                       float* __restrict__ pg) {
  int r = blockIdx.x * blockDim.x + threadIdx.x;
  if (r >= Rc) return;
  const int D = 2 * Hc + Ec;  // 900, row stride of pgen_W
  const float* c  = concat + (long)r * 2 * Hc;
  const float* xr = x + (long)r * Ec;
  float z0 = pb[0], z1 = pb[1];
  for (int k = 0; k < 2 * Hc; ++k) {
    z0 = fmaf(pw[k], c[k], z0);
    z1 = fmaf(pw[D + k], c[k], z1);
  }
  for (int k = 0; k < Ec; ++k) {
    z0 = fmaf(pw[2 * Hc + k], xr[k], z0);
    z1 = fmaf(pw[D + 2 * Hc + k], xr[k], z1);
  }
  float m = fmaxf(z0, z1);
  float e0 = __expf(z0 - m), e1 = __expf(z1 - m);
  float s = e0 + e1;
  pg[r * 2 + 0] = e0 / s;
  pg[r * 2 + 1] = e1 / s;
}

// ---------------------------------------------------------------------------
// K8: per-row max & sum(exp) over the 50000 logits living in d_out.
// ---------------------------------------------------------------------------
__global__ void k_rowstats(const float* __restrict__ logits,
                           float* __restrict__ stats) {
  __shared__ float red[256];
  int r = blockIdx.x, tid = threadIdx.x;
  const float* row = logits + (long)r * VPc;
  float mx = -INFINITY;
  for (int v = tid; v < Vc; v += 256) mx = fmaxf(mx, row[v]);
  red[tid] = mx;
  __syncthreads();
  for (int o = 128; o > 0; o >>= 1) {
    if (tid < o) red[tid] = fmaxf(red[tid], red[tid + o]);
    __syncthreads();
  }
  mx = red[0];
  __syncthreads();
  float sm = 0.f;
  for (int v = tid; v < Vc; v += 256) sm += __expf(row[v] - mx);
  red[tid] = sm;
  __syncthreads();
  for (int o = 128; o > 0; o >>= 1) {
    if (tid < o) red[tid] += red[tid + o];
    __syncthreads();
  }
  if (tid == 0) {
    stats[r * 2 + 0] = mx;
    stats[r * 2 + 1] = red[0];
  }
}

// ---------------------------------------------------------------------------
// K9: in-place logits -> p_gen0 * softmax; zero the OOV pad slots.
// ---------------------------------------------------------------------------
__global__ void k_finalize(float* __restrict__ out,
                           const float* __restrict__ stats,
                           const float* __restrict__ pg) {
  long idx = (long)blockIdx.x * blockDim.x + threadIdx.x;
  const long tot = (long)Rc * VPc;
  if (idx >= tot) return;
  int r = (int)(idx / VPc);
  int v = (int)(idx % VPc);
  float val = 0.f;
  if (v < Vc) {
    val = pg[r * 2] * __expf(out[idx] - stats[r * 2]) / stats[r * 2 + 1];
  }
  out[idx] = val;
}

// ---------------------------------------------------------------------------
// K10: copy-mechanism scatter-add of p_gen1 * attn into final dist.
// ---------------------------------------------------------------------------
__global__ void k_scatter(float* __restrict__ out,
                          const float* __restrict__ attn,
                          const int* __restrict__ src_oov,
                          const float* __restrict__ pg) {
  int idx = blockIdx.x * blockDim.x + threadIdx.x;
  if (idx >= Rc * Sc) return;
  int r = idx / Sc, s = idx % Sc;
  int b = r / Tc;
  int v = src_oov[b * Sc + s];
  atomicAdd(out + (long)r * VPc + v, pg[r * 2 + 1] * attn[idx]);
}

// ---------------------------------------------------------------------------
extern "C" void kernel_launch(void* const* d_in, const int* in_sizes, int n_in,
                              void* d_out, int out_size, void* d_ws, size_t ws_size,
                              hipStream_t stream) {
  const int*           tokens  = (const int*)d_in[0];
  const float*         enc     = (const float*)d_in[1];
  const unsigned char* mask    = (const unsigned char*)d_in[2];
  const float*         hidden0 = (const float*)d_in[3];
  const int*           src_oov = (const int*)d_in[4];
  // d_in[5] = max_num_oov (compile-time constant here)
  const float* embed = (const float*)d_in[6];
  const float* Wih   = (const float*)d_in[7];
  const float* Whh   = (const float*)d_in[8];
  const float* bih   = (const float*)d_in[9];
  const float* bhh   = (const float*)d_in[10];
  const float* attnW = (const float*)d_in[11];
  const float* attnB = (const float*)d_in[12];
  const float* fcW   = (const float*)d_in[13];
  const float* fcB   = (const float*)d_in[14];
  const float* outW  = (const float*)d_in[15];
  const float* outB  = (const float*)d_in[16];
  const float* pgW   = (const float*)d_in[17];
  const float* pgB   = (const float*)d_in[18];

  // workspace layout (~3.9 MB of floats)
  float* ws      = (float*)d_ws;
  float* x       = ws;                    // R*E
  float* hiddens = x + Rc * Ec;           // R*H
  float* q       = hiddens + Rc * Hc;     // R*H
  float* concat  = q + Rc * Hc;           // R*2H
  float* feat    = concat + Rc * 2 * Hc;  // R*H
  float* hA      = feat + Rc * Hc;        // L*B*H
  float* hB      = hA + Lc * Bc * Hc;     // L*B*H
  float* pg      = hB + Lc * Bc * Hc;     // R*2
  float* stats   = pg + Rc * 2;           // R*2

  float* fout     = (float*)d_out;            // [R, VPc] final_dist
  float* attn_out = fout + (long)Rc * VPc;    // [R, S]   attn_dist

  // 1. embedding
  k_embed<<<(Rc * Ec + 255) / 256, 256, 0, stream>>>(tokens, embed, x);

  // 2. GRU scan (ping-pong hidden state; never mutate hidden0 input)
  hipMemcpyAsync(hA, hidden0, (size_t)Lc * Bc * Hc * sizeof(float),
                 hipMemcpyDeviceToDevice, stream);
  float* hin = hA;
  float* hout = hB;
  for (int t = 0; t < Tc; ++t) {
    for (int l = 0; l < Lc; ++l) {
      const float* inp;
      int istride;
      if (l == 0) { inp = x + (long)t * Ec; istride = Tc * Ec; }
      else        { inp = hout + (long)(l - 1) * Bc * Hc; istride = Hc; }
      k_gru_cell<<<Bc, 320, 0, stream>>>(
          inp, istride, hin + (long)l * Bc * Hc, hout + (long)l * Bc * Hc,
          Wih + (long)l * 3 * Hc * Ec, Whh + (long)l * 3 * Hc * Hc,
          bih + (long)l * 3 * Hc, bhh + (long)l * 3 * Hc,
          (l == Lc - 1) ? (hiddens + (long)t * Hc) : nullptr);
    }
    float* tmp = hin; hin = hout; hout = tmp;
  }

  // 3. q = hiddens @ attn_W.T + attn_b   (WMMA, M=512 N=300 K=300)
  {
    int blocks_m = Rc / 32;                 // 16
    int blocks_n = (Hc + 255) / 256;        // 2
    k_gemm_nt_wmma<<<blocks_m * blocks_n, 256, 0, stream>>>(
        hiddens, Hc, attnW, Hc, attnB, q, (long)Hc, blocks_m, Hc, Hc);
  }

  // 4. attention softmax + contexts (writes attn_dist output + concat buffer)
  k_attn<<<Rc, 256, 0, stream>>>(q, enc, mask, hiddens, attn_out, concat);

  // 5. feat = [ctx, hid] @ fc_W.T + fc_b   (WMMA, K=600 -> 2 LDS chunks)
  {
    int blocks_m = Rc / 32;
    int blocks_n = (Hc + 255) / 256;
    k_gemm_nt_wmma<<<blocks_m * blocks_n, 256, 0, stream>>>(
        concat, 2 * Hc, fcW, 2 * Hc, fcB, feat, (long)Hc, blocks_m, Hc, 2 * Hc);
  }

  // 6. p_gen
  k_pgen<<<(Rc + 255) / 256, 256, 0, stream>>>(concat, x, pgW, pgB, pg);

  // 7. logits = feat @ out_W.T + out_b written straight into d_out rows
  //    (WMMA, M=512 N=50000 K=300, row stride VPc)
  {
    int blocks_m = Rc / 32;                 // 16
    int blocks_n = (Vc + 255) / 256;        // 196
    k_gemm_nt_wmma<<<blocks_m * blocks_n, 256, 0, stream>>>(
        feat, Hc, outW, Hc, outB, fout, (long)VPc, blocks_m, Vc, Hc);
  }

  // 8./9. softmax over vocab (two streaming passes) scaled by p_gen0
  k_rowstats<<<Rc, 256, 0, stream>>>(fout, stats);
  {
    long tot = (long)Rc * VPc;
    k_finalize<<<(int)((tot + 255) / 256), 256, 0, stream>>>(fout, stats, pg);
  }

  // 10. copy mechanism scatter-add
  k_scatter<<<(Rc * Sc + 255) / 256, 256, 0, stream>>>(fout, attn_out, src_oov, pg);
}